// MyLayer_56478819942817
// MI455X (gfx1250) — compile-verified
//
#include <hip/hip_runtime.h>
#include <hip/hip_bf16.h>

typedef __attribute__((ext_vector_type(2))) float v2f;
typedef __attribute__((ext_vector_type(8))) float v8f;

__device__ __forceinline__ float lrelu(float v) {
    return (v >= 0.0f) ? v : 0.01f * v;
}

__device__ __forceinline__ void atomic_add_f32(float* p, float v) {
    __hip_atomic_fetch_add(p, v, __ATOMIC_RELAXED, __HIP_MEMORY_SCOPE_AGENT);
}

// D(16x16) += A(16x4) * B(4x16), fp32 WMMA (matches reference f32 math).
__device__ __forceinline__ v8f wmma4(v2f a, v2f b, v8f c) {
#if __has_builtin(__builtin_amdgcn_wmma_f32_16x16x4_f32)
    return __builtin_amdgcn_wmma_f32_16x16x4_f32(
        /*neg_a=*/false, a, /*neg_b=*/false, b,
        /*c_mod=*/(short)0, c, /*reuse_a=*/false, /*reuse_b=*/false);
#else
    // Cross-lane scalar fallback (compile-safety only).
    const int lane   = threadIdx.x & 31;
    const int lane16 = lane & 15;
    const int half   = lane >> 4;
#pragma unroll
    for (int v = 0; v < 8; ++v) {
        const int M = half * 8 + v;
        const int N = lane16;
        float a0 = __shfl(a.x, M, 32),      a1 = __shfl(a.y, M, 32);
        float a2 = __shfl(a.x, M + 16, 32), a3 = __shfl(a.y, M + 16, 32);
        float b0 = __shfl(b.x, N, 32),      b1 = __shfl(b.y, N, 32);
        float b2 = __shfl(b.x, N + 16, 32), b3 = __shfl(b.y, N + 16, 32);
        c[v] += a0 * b0 + a1 * b1 + a2 * b2 + a3 * b3;
    }
    return c;
#endif
}

__global__ void fill_f32(float* __restrict__ p, long n, float v) {
    long i = (long)blockIdx.x * blockDim.x + threadIdx.x;
    if (i < n) p[i] = v;
}

__global__ void edge_count(const int* __restrict__ edge, float* __restrict__ cnt, int E) {
    int e = blockIdx.x * blockDim.x + threadIdx.x;
    if (e < E) atomic_add_f32(&cnt[edge[e]], 1.0f);
}

__global__ void inv_count(const float* __restrict__ cnt, float* __restrict__ inv, int N) {
    int i = blockIdx.x * blockDim.x + threadIdx.x;
    if (i < N) {
        float c = cnt[i];
        inv[i] = (c > 0.0f) ? (1.0f / c) : 0.0f;
    }
}

// Per-edge: msg = lrelu(W * (h[col]*w) + b), atomically accumulated into agg[row].
// One wave handles 16 edges; the FIN-wide contraction is done as FIN/4 chunks of
// V_WMMA_F32_16X16X4_F32, with FOUT covered by ceil(FOUT/16) M-tiles.
template <int FIN, int FOUT>
__global__ void gnn_edge_wmma(const float* __restrict__ h,
                              const int* __restrict__ edge,
                              const float* __restrict__ ew,
                              const float* __restrict__ W,
                              const float* __restrict__ bias,
                              float* __restrict__ agg,
                              int E) {
    constexpr int KC = FIN / 4;
    constexpr int MT = (FOUT + 15) / 16;

    const int lane   = threadIdx.x & 31;
    const int lane16 = lane & 15;
    const int half   = lane >> 4;

    const long wave = (long)blockIdx.x * (blockDim.x >> 5) + (threadIdx.x >> 5);
    const long e    = wave * 16 + lane16;
    const bool ev   = e < (long)E;
    const long ec   = ev ? e : 0;

    const int   r = edge[ec];            // destination row (scatter target)
    const int   c = edge[(long)E + ec];  // source col (gather index)
    const float w = ev ? ew[ec] : 0.0f;

    // B operand: 16 columns = 16 edges' scaled features. Component v of chunk kc
    // holds K = kc*4 + half*2 + v for column N = lane16.
    v2f bm[KC];
#pragma unroll
    for (int kc = 0; kc < KC; ++kc) {
        const float* hp = h + (long)c * FIN + kc * 4 + half * 2;
        bm[kc].x = hp[0] * w;
        bm[kc].y = hp[1] * w;
    }

#pragma unroll
    for (int t = 0; t < MT; ++t) {
        v8f acc = {0.f, 0.f, 0.f, 0.f, 0.f, 0.f, 0.f, 0.f};
#pragma unroll
        for (int kc = 0; kc < KC; ++kc) {
            const int m = t * 16 + lane16;  // A row M = lane16
            v2f am;
            if (m < FOUT) {
                const float* wp = W + (long)m * FIN + kc * 4 + half * 2;
                am.x = wp[0];
                am.y = wp[1];
            } else {
                am.x = 0.0f;
                am.y = 0.0f;
            }
            acc = wmma4(am, bm[kc], acc);  // EXEC is all-1s here
        }
        // D: VGPR v holds M = t*16 + half*8 + v for edge N = lane16.
#pragma unroll
        for (int v = 0; v < 8; ++v) {
            const int m = t * 16 + half * 8 + v;
            if (m < FOUT && ev) {
                float val = lrelu(acc[v] + bias[m]);
                atomic_add_f32(&agg[(long)r * FOUT + m], val);
            }
        }
    }
}

// out = agg*inv_cnt + lrelu(h @ W.T + b)  (residual branch + scatter-mean scale)
template <int FIN, int FOUT>
__global__ void gnn_node(const float* __restrict__ h,
                         const float* __restrict__ W,
                         const float* __restrict__ bias,
                         const float* __restrict__ agg,
                         const float* __restrict__ inv,
                         float* __restrict__ out,
                         int N) {
    int n = blockIdx.x * blockDim.x + threadIdx.x;
    if (n >= N) return;
    float hv[FIN];
#pragma unroll
    for (int k = 0; k < FIN; ++k) hv[k] = h[(long)n * FIN + k];
    const float s_inv = inv[n];
#pragma unroll
    for (int m = 0; m < FOUT; ++m) {
        float s = bias[m];
#pragma unroll
        for (int k = 0; k < FIN; ++k) s += W[m * FIN + k] * hv[k];
        out[(long)n * FOUT + m] = agg[(long)n * FOUT + m] * s_inv + lrelu(s);
    }
}

// Hierarchical column-sum of h3[N,32] into pooled[32].
__global__ void pool_kernel(const float* __restrict__ h3, float* __restrict__ pooled, int N) {
    __shared__ float sm[256];
    const int tid = threadIdx.x;
    const int f   = tid & 31;
    const int grp = tid >> 5;  // 8 node-groups per block
    const long g0 = (long)blockIdx.x * 8 + grp;
    const long gs = (long)gridDim.x * 8;
    float acc = 0.0f;
    for (long n = g0; n < (long)N; n += gs) acc += h3[n * 32 + f];
    sm[tid] = acc;
    __syncthreads();
    if (tid < 32) {
        float s = 0.0f;
#pragma unroll
        for (int g = 0; g < 8; ++g) s += sm[g * 32 + tid];
        atomic_add_f32(&pooled[tid], s);
    }
}

// logits = (pooled/N) @ W7.T + b7 ; out = log_softmax(logits)
__global__ void head_kernel(const float* __restrict__ pooled,
                            const float* __restrict__ W7,
                            const float* __restrict__ b7,
                            float* __restrict__ out,
                            float invN) {
    if (threadIdx.x == 0 && blockIdx.x == 0) {
        float l0 = b7[0], l1 = b7[1];
        for (int f = 0; f < 32; ++f) {
            float m = pooled[f] * invN;
            l0 += W7[f] * m;
            l1 += W7[32 + f] * m;
        }
        float mx  = fmaxf(l0, l1);
        float lse = mx + logf(expf(l0 - mx) + expf(l1 - mx));
        out[0] = l0 - lse;
        out[1] = l1 - lse;
    }
}

extern "C" void kernel_launch(void* const* d_in, const int* in_sizes, int n_in,
                              void* d_out, int out_size, void* d_ws, size_t ws_size,
                              hipStream_t stream) {
    const float* x    = (const float*)d_in[0];
    const int*   edge = (const int*)d_in[1];
    const float* wgt  = (const float*)d_in[2];
    const float* W1   = (const float*)d_in[3];
    const float* b1   = (const float*)d_in[4];
    const float* W3   = (const float*)d_in[5];
    const float* b3   = (const float*)d_in[6];
    const float* W5   = (const float*)d_in[7];
    const float* b5   = (const float*)d_in[8];
    const float* W7   = (const float*)d_in[9];
    const float* b7   = (const float*)d_in[10];
    float* out = (float*)d_out;

    const int N = in_sizes[0] / 4;  // 100000 nodes
    const int E = in_sizes[2];      // 3200000 edges

    // Workspace carve-out (256B aligned): ~36 MB total, all L2-resident.
    char* ws = (char*)d_ws;
    auto carve = [&](size_t bytes) {
        char* p = ws;
        ws += (bytes + 255) & ~(size_t)255;
        return (float*)p;
    };
    float* cnt    = carve((size_t)N * 4);
    float* inv    = carve((size_t)N * 4);
    float* h1     = carve((size_t)N * 8 * 4);
    float* h2     = carve((size_t)N * 16 * 4);
    float* h3     = carve((size_t)N * 32 * 4);
    float* agg    = carve((size_t)N * 32 * 4);  // reused per block
    float* pooled = carve(32 * 4);

    const int TB = 256;
    const int nodeBlk = (N + TB - 1) / TB;
    const int edgeBlk = (E + TB - 1) / TB;
    const long ewaves = ((long)E + 15) / 16;
    const int  ewBlk  = (int)((ewaves + 7) / 8);  // 8 waves / 256-thread block

    // Per-row edge counts -> inv_cnt (scatter_mean denominators).
    fill_f32<<<nodeBlk, TB, 0, stream>>>(cnt, N, 0.0f);
    edge_count<<<edgeBlk, TB, 0, stream>>>(edge, cnt, E);
    inv_count<<<nodeBlk, TB, 0, stream>>>(cnt, inv, N);

    // Block 1: 4 -> 8
    fill_f32<<<((long)N * 8 + TB - 1) / TB, TB, 0, stream>>>(agg, (long)N * 8, 0.0f);
    gnn_edge_wmma<4, 8><<<ewBlk, TB, 0, stream>>>(x, edge, wgt, W1, b1, agg, E);
    gnn_node<4, 8><<<nodeBlk, TB, 0, stream>>>(x, W1, b1, agg, inv, h1, N);

    // Block 2: 8 -> 16
    fill_f32<<<((long)N * 16 + TB - 1) / TB, TB, 0, stream>>>(agg, (long)N * 16, 0.0f);
    gnn_edge_wmma<8, 16><<<ewBlk, TB, 0, stream>>>(h1, edge, wgt, W3, b3, agg, E);
    gnn_node<8, 16><<<nodeBlk, TB, 0, stream>>>(h1, W3, b3, agg, inv, h2, N);

    // Block 3: 16 -> 32
    fill_f32<<<((long)N * 32 + TB - 1) / TB, TB, 0, stream>>>(agg, (long)N * 32, 0.0f);
    gnn_edge_wmma<16, 32><<<ewBlk, TB, 0, stream>>>(h2, edge, wgt, W5, b5, agg, E);
    gnn_node<16, 32><<<nodeBlk, TB, 0, stream>>>(h2, W5, b5, agg, inv, h3, N);

    // Global mean pool + classifier head + log_softmax.
    fill_f32<<<1, 32, 0, stream>>>(pooled, 32, 0.0f);
    pool_kernel<<<128, TB, 0, stream>>>(h3, pooled, N);
    head_kernel<<<1, 32, 0, stream>>>(pooled, W7, b7, out, 1.0f / (float)N);
}